// LocalGrouper_3015067041923
// MI455X (gfx1250) — compile-verified
//
#include <hip/hip_runtime.h>

#define NPTS   8192
#define BATCH  16
#define SGRP   1024
#define KNN    24
#define DCH    64

typedef float v2f __attribute__((ext_vector_type(2)));
typedef float v8f __attribute__((ext_vector_type(8)));

// ---------------------------------------------------------------------------
// Phase 1: Farthest Point Sampling. One block per batch, 1024 threads,
// each thread owns 8 points (coords + running min-distance in registers).
// Sequential 1024-step chain; per step: register update + shuffle argmax +
// LDS cross-wave reduction. Deterministic start at index 0 (matches ref).
// ---------------------------------------------------------------------------
__global__ __launch_bounds__(1024) void fps_kernel(const float* __restrict__ xyz,
                                                   int* __restrict__ fps_idx) {
    const int b = blockIdx.x;
    const int t = threadIdx.x;
    const float* xb = xyz + (size_t)b * NPTS * 3;

    float px[8], py[8], pz[8], dist[8];
#pragma unroll
    for (int j = 0; j < 8; ++j) {
        int p = t + 1024 * j;
        px[j] = xb[p * 3 + 0];
        py[j] = xb[p * 3 + 1];
        pz[j] = xb[p * 3 + 2];
        dist[j] = 1e10f;
    }

    __shared__ float s_val[32];
    __shared__ int   s_idx[32];
    __shared__ int   s_win;

    int cur = 0;
    if (t == 0) fps_idx[b * SGRP + 0] = 0;

    for (int it = 1; it < SGRP; ++it) {
        const float cx = xb[cur * 3 + 0];
        const float cy = xb[cur * 3 + 1];
        const float cz = xb[cur * 3 + 2];

        float best = -1.0f;
        int   bidx = 0;
#pragma unroll
        for (int j = 0; j < 8; ++j) {
            float dx = px[j] - cx, dy = py[j] - cy, dz = pz[j] - cz;
            float d = dx * dx + dy * dy + dz * dz;
            dist[j] = fminf(dist[j], d);
            if (dist[j] > best) { best = dist[j]; bidx = t + 1024 * j; }
        }
        // wave32 argmax reduction (tie -> smaller index, matching jnp.argmax)
#pragma unroll
        for (int off = 16; off > 0; off >>= 1) {
            float ov = __shfl_down(best, off, 32);
            int   oi = __shfl_down(bidx, off, 32);
            if (ov > best || (ov == best && oi < bidx)) { best = ov; bidx = oi; }
        }
        const int wave = t >> 5;
        if ((t & 31) == 0) { s_val[wave] = best; s_idx[wave] = bidx; }
        __syncthreads();
        if (t == 0) {
            float bv = s_val[0]; int bi = s_idx[0];
            for (int w = 1; w < 32; ++w) {
                if (s_val[w] > bv || (s_val[w] == bv && s_idx[w] < bi)) {
                    bv = s_val[w]; bi = s_idx[w];
                }
            }
            s_win = bi;
            fps_idx[b * SGRP + it] = bi;
        }
        __syncthreads();
        cur = s_win;
        __syncthreads();
    }
}

// ---------------------------------------------------------------------------
// Phase 2: KNN. One wave (32 threads) per 16-query tile; 16*64 = 1024 blocks.
// Distances via V_WMMA_F32_16X16X4_F32:
//   A row m  (point)  = ( px, py, pz, |p|^2 )
//   B col n  (query)  = (-2qx,-2qy,-2qz, 1  )
//   D[m][n] + |q_n|^2 = full squared distance (f32, exact vs reference)
// Lane layout: col n = lane&15, half = lane>>4 owns rows r+8*half.
// Each lane keeps a sorted top-24 (dist,idx) list in registers; the two
// lanes per query are merged through LDS at the end (sorted, ties->low idx).
// ---------------------------------------------------------------------------
__global__ __launch_bounds__(32) void knn_kernel(const float* __restrict__ xyz,
                                                 const int* __restrict__ fps_idx,
                                                 int* __restrict__ knn_idx) {
    const int b    = blockIdx.x >> 6;    // 64 tiles of 16 queries per batch
    const int tile = blockIdx.x & 63;
    const int lane = threadIdx.x;
    const int col  = lane & 15;
    const int half = lane >> 4;
    const float* xb = xyz + (size_t)b * NPTS * 3;

    // B fragment: this lane's query column
    const int s  = tile * 16 + col;
    const int qi = fps_idx[b * SGRP + s];
    const float qx = xb[qi * 3 + 0];
    const float qy = xb[qi * 3 + 1];
    const float qz = xb[qi * 3 + 2];
    const float qn = qx * qx + qy * qy + qz * qz;
    v2f bf;
    if (half == 0) { bf.x = -2.0f * qx; bf.y = -2.0f * qy; }
    else           { bf.x = -2.0f * qz; bf.y = 1.0f; }

    float bd[KNN];
    int   bi[KNN];
#pragma unroll
    for (int j = 0; j < KNN; ++j) { bd[j] = 3.4e38f; bi[j] = 0; }

    for (int n0 = 0; n0 < NPTS; n0 += 16) {
        // A fragment: point tile (each lane loads row m = n0 + col)
        const int m = n0 + col;
        const float fx = xb[m * 3 + 0];
        const float fy = xb[m * 3 + 1];
        const float fz = xb[m * 3 + 2];
        const float pn = fx * fx + fy * fy + fz * fz;
        v2f a;
        if (half == 0) { a.x = fx; a.y = fy; }
        else           { a.x = fz; a.y = pn; }

        v8f c = {};
        c = __builtin_amdgcn_wmma_f32_16x16x4_f32(
                /*neg_a=*/false, a, /*neg_b=*/false, bf,
                /*c_mod=*/(short)0, c, /*reuse_a=*/false, /*reuse_b=*/false);

#pragma unroll
        for (int r = 0; r < 8; ++r) {
            const float d   = c[r] + qn;
            const int  pidx = n0 + r + 8 * half;
            if (d < bd[KNN - 1]) {
                bd[KNN - 1] = d;
                bi[KNN - 1] = pidx;
#pragma unroll
                for (int j = KNN - 1; j > 0; --j) {
                    if (bd[j] < bd[j - 1]) {
                        float tf = bd[j]; bd[j] = bd[j - 1]; bd[j - 1] = tf;
                        int   ti = bi[j]; bi[j] = bi[j - 1]; bi[j - 1] = ti;
                    }
                }
            }
        }
    }

    // Merge the two per-query lanes via LDS (both lists already sorted).
    __shared__ float sd[32][KNN];
    __shared__ int   si[32][KNN];
#pragma unroll
    for (int j = 0; j < KNN; ++j) { sd[lane][j] = bd[j]; si[lane][j] = bi[j]; }
    __syncthreads();

    if (half == 0) {
        int i0 = 0, i1 = 0;
        int* out = knn_idx + ((size_t)b * SGRP + s) * KNN;
        for (int j = 0; j < KNN; ++j) {
            const float d0 = sd[lane][i0],      d1 = sd[lane + 16][i1];
            const int   x0 = si[lane][i0],      x1 = si[lane + 16][i1];
            const bool take0 = (d0 < d1) || (d0 == d1 && x0 < x1);
            out[j] = take0 ? x0 : x1;
            if (take0) ++i0; else ++i1;
        }
    }
}

// ---------------------------------------------------------------------------
// Phase 3: gathers (pure bandwidth).
// ---------------------------------------------------------------------------
__global__ void gather_new_kernel(const float* __restrict__ xyz,
                                  const float* __restrict__ points,
                                  const int* __restrict__ fps_idx,
                                  float* __restrict__ new_xyz,
                                  float* __restrict__ new_points) {
    const int i = blockIdx.x * blockDim.x + threadIdx.x;   // over B*S
    if (i >= BATCH * SGRP) return;
    const int b = i / SGRP;
    const int p = fps_idx[i];
    const float* xs = xyz + ((size_t)b * NPTS + p) * 3;
    float* xd = new_xyz + (size_t)i * 3;
    xd[0] = xs[0]; xd[1] = xs[1]; xd[2] = xs[2];
    const float* ps = points + ((size_t)b * NPTS + p) * DCH;
    float* pd = new_points + (size_t)i * DCH;
#pragma unroll 4
    for (int c = 0; c < DCH; ++c) pd[c] = ps[c];
}

__global__ void gather_grouped_kernel(const float* __restrict__ xyz,
                                      const float* __restrict__ points,
                                      const int* __restrict__ knn_idx,
                                      float* __restrict__ grouped_xyz,
                                      float* __restrict__ grouped_points) {
    const int i = blockIdx.x * blockDim.x + threadIdx.x;   // over B*S*K
    if (i >= BATCH * SGRP * KNN) return;
    const int b = i / (SGRP * KNN);
    const int p = knn_idx[i];
    const float* xs = xyz + ((size_t)b * NPTS + p) * 3;
    float* xd = grouped_xyz + (size_t)i * 3;
    xd[0] = xs[0]; xd[1] = xs[1]; xd[2] = xs[2];
    const float* ps = points + ((size_t)b * NPTS + p) * DCH;
    float* pd = grouped_points + (size_t)i * DCH;
#pragma unroll 4
    for (int c = 0; c < DCH; ++c) pd[c] = ps[c];
}

// ---------------------------------------------------------------------------
extern "C" void kernel_launch(void* const* d_in, const int* in_sizes, int n_in,
                              void* d_out, int out_size, void* d_ws, size_t ws_size,
                              hipStream_t stream) {
    const float* xyz    = (const float*)d_in[0];   // [16, 8192, 3]
    const float* points = (const float*)d_in[1];   // [16, 8192, 64]

    float* out            = (float*)d_out;
    float* new_xyz        = out;                                       // [16,1024,3]
    float* new_points     = new_xyz + (size_t)BATCH * SGRP * 3;        // [16,1024,64]
    float* grouped_xyz    = new_points + (size_t)BATCH * SGRP * DCH;   // [16,1024,24,3]
    float* grouped_points = grouped_xyz + (size_t)BATCH * SGRP * KNN * 3;

    int* fps_idx = (int*)d_ws;                       // [16,1024]
    int* knn_idx = fps_idx + BATCH * SGRP;           // [16,1024,24]

    fps_kernel<<<BATCH, 1024, 0, stream>>>(xyz, fps_idx);
    knn_kernel<<<BATCH * (SGRP / 16), 32, 0, stream>>>(xyz, fps_idx, knn_idx);
    gather_new_kernel<<<(BATCH * SGRP + 255) / 256, 256, 0, stream>>>(
        xyz, points, fps_idx, new_xyz, new_points);
    gather_grouped_kernel<<<(BATCH * SGRP * KNN + 255) / 256, 256, 0, stream>>>(
        xyz, points, knn_idx, grouped_xyz, grouped_points);
}